// NLLLoss_64407329571149
// MI455X (gfx1250) — compile-verified
//
#include <hip/hip_runtime.h>

// Gaussian NLL pieces: per-element 3x3 SPD solve + three global means.
// Memory-bound streaming reduction: 192 MB in / 12 B out -> ~8.2 us floor @ 23.3 TB/s.
// Stage 1 (hot): grid-stride f32 element math, NT-hinted wide loads, f64
//   accumulators, wave32 ds_bpermute shuffle reduction, deterministic block
//   partials to d_ws.
// Stage 2 (fan-in): WMMA ones-matrix reduction (v_wmma_f32_16x16x4_f32):
//   B=ones => D rows are row-sums of A; summing D's 8 VGPRs + one lane-16
//   shuffle gives the full 32-lane sum in one matrix op per scalar.

#define GNLL_EPS 1e-3f
#define GNLL_BLOCK 256
#define GNLL_MAXBLOCKS 2048

typedef float v2f __attribute__((ext_vector_type(2)));
typedef float v8f __attribute__((ext_vector_type(8)));

__device__ __forceinline__ double waveReduceAdd(double v) {
#pragma unroll
  for (int off = 16; off > 0; off >>= 1)
    v += __shfl_down(v, off, 32);   // wave32: ds_bpermute path
  return v;
}

// Full-wave (32 lane) f32 sum via the matrix unit.
// A VGPR0 = x, VGPR1 = 0; B = ones(4x16). D[m][n] = sum of row m's A slots.
// Rows partition the 64 A slots, so Sum_m rowsum[m] == Sum_lanes x.
// D layout: VGPR r -> M=r (lanes 0-15), M=r+8 (lanes 16-31); summing the 8
// D VGPRs gives rowsums 0-7 in lanes 0-15 and 8-15 in lanes 16-31; one
// shuffle-by-16 combines them. EXEC must be all ones at the call site.
__device__ __forceinline__ float waveWmmaSum(float x) {
  v2f a; a[0] = x;    a[1] = 0.0f;
  v2f b; b[0] = 1.0f; b[1] = 1.0f;
  v8f c = {0.0f, 0.0f, 0.0f, 0.0f, 0.0f, 0.0f, 0.0f, 0.0f};
  v8f d = __builtin_amdgcn_wmma_f32_16x16x4_f32(
      /*neg_a=*/false, a, /*neg_b=*/false, b,
      /*c_mod=*/(short)0, c, /*reuse_a=*/false, /*reuse_b=*/false);
  float s = ((d[0] + d[1]) + (d[2] + d[3])) + ((d[4] + d[5]) + (d[6] + d[7]));
  s += __shfl_down(s, 16, 32);
  return s;   // valid in lane 0
}

// Block-level reduce of 3 doubles (shuffle+LDS, f64); result valid on thread 0.
__device__ __forceinline__ void blockReduce3(double& sq, double& sl, double& sv) {
  __shared__ double lsq[GNLL_BLOCK / 32];
  __shared__ double lsl[GNLL_BLOCK / 32];
  __shared__ double lsv[GNLL_BLOCK / 32];
  const int lane = threadIdx.x & 31;
  const int wave = threadIdx.x >> 5;
  sq = waveReduceAdd(sq);
  sl = waveReduceAdd(sl);
  sv = waveReduceAdd(sv);
  if (lane == 0) { lsq[wave] = sq; lsl[wave] = sl; lsv[wave] = sv; }
  __syncthreads();
  if (wave == 0) {
    const int nw = GNLL_BLOCK / 32;
    double a = (lane < nw) ? lsq[lane] : 0.0;
    double b = (lane < nw) ? lsl[lane] : 0.0;
    double c = (lane < nw) ? lsv[lane] : 0.0;
    sq = waveReduceAdd(a);
    sl = waveReduceAdd(b);
    sv = waveReduceAdd(c);
  }
}

__global__ __launch_bounds__(GNLL_BLOCK) void gnll_partial_kernel(
    const float* __restrict__ prd, const float* __restrict__ tgt,
    const float* __restrict__ fv, double* __restrict__ partials, int n) {
  const int tid = blockIdx.x * GNLL_BLOCK + threadIdx.x;
  const int stride = gridDim.x * GNLL_BLOCK;

  double sq = 0.0, sl = 0.0, sv = 0.0;

  for (int i = tid; i < n; i += stride) {
    const float* p3 = prd + 3 * i;
    const float* t3 = tgt + 3 * i;
    const v2f* l2 = (const v2f*)(fv + 6 * i);   // 24*i bytes: 8B aligned

    // Streaming loads, non-temporal (single-use data; keep L2 clean).
    v2f l01 = __builtin_nontemporal_load(l2 + 0);
    v2f l23 = __builtin_nontemporal_load(l2 + 1);
    v2f l45 = __builtin_nontemporal_load(l2 + 2);
    float d0 = __builtin_nontemporal_load(t3 + 0) - __builtin_nontemporal_load(p3 + 0);
    float d1 = __builtin_nontemporal_load(t3 + 1) - __builtin_nontemporal_load(p3 + 1);
    float d2 = __builtin_nontemporal_load(t3 + 2) - __builtin_nontemporal_load(p3 + 2);

    // L = [[a,0,0],[f,b,0],[e,d,c]]  (a=l0 b=l1 c=l2 d=l3 e=l4 f=l5)
    const float a = l01[0], b = l01[1], c = l23[0], d = l23[1], e = l45[0], f = l45[1];

    // cov = L*L^T + eps*I (symmetric: A B C / B D E / C E F)
    const float A = fmaf(a, a, GNLL_EPS);
    const float B = a * f;
    const float C = a * e;
    const float D = fmaf(f, f, fmaf(b, b, GNLL_EPS));
    const float E = fmaf(f, e, b * d);
    const float F = fmaf(e, e, fmaf(d, d, fmaf(c, c, GNLL_EPS)));

    // Adjugate (symmetric) and determinant.
    const float i00 = fmaf(D, F, -E * E);
    const float i01 = fmaf(C, E, -B * F);
    const float i02 = fmaf(B, E, -C * D);
    const float i11 = fmaf(A, F, -C * C);
    const float i12 = fmaf(B, C, -A * E);
    const float i22 = fmaf(A, D, -B * B);
    const float det = fmaf(A, i00, fmaf(B, i01, C * i02));   // > 0 (SPD)

    // diff^T cov^{-1} diff = (diff^T adj diff) / det
    const float qnum =
        fmaf(d0 * d0, i00,
        fmaf(d1 * d1, i11,
        fmaf(d2 * d2, i22,
        2.0f * fmaf(d0 * d1, i01,
               fmaf(d0 * d2, i02, d1 * d2 * i12)))));

    sq += (double)(qnum / det);
    sl += (double)logf(det);                 // slogdet: sign==+1 for SPD
    sv += (double)(((a + b) + (c + d)) + (e + f));
  }

  blockReduce3(sq, sl, sv);
  if (threadIdx.x == 0) {
    partials[3 * blockIdx.x + 0] = sq;
    partials[3 * blockIdx.x + 1] = sl;
    partials[3 * blockIdx.x + 2] = sv;
  }
}

__global__ __launch_bounds__(GNLL_BLOCK) void gnll_final_kernel(
    const double* __restrict__ partials, int nparts, float* __restrict__ out,
    double invN) {
  // Per-thread f64 accumulation over strided block partials.
  double sq = 0.0, sl = 0.0, sv = 0.0;
  for (int i = threadIdx.x; i < nparts; i += GNLL_BLOCK) {
    sq += partials[3 * i + 0];
    sl += partials[3 * i + 1];
    sv += partials[3 * i + 2];
  }
  // Pre-scale to O(1) magnitudes, then matrix-unit fan-in in f32.
  float xq = (float)(sq * invN);
  float xl = (float)(sl * invN);
  float xv = (float)(sv * invN * (1.0 / 6.0));

  __shared__ float wq[GNLL_BLOCK / 32];
  __shared__ float wl[GNLL_BLOCK / 32];
  __shared__ float wv[GNLL_BLOCK / 32];
  const int lane = threadIdx.x & 31;
  const int wave = threadIdx.x >> 5;

  // Stage A: per-wave WMMA reduction (full EXEC: no divergence yet).
  float rq = waveWmmaSum(xq);
  float rl = waveWmmaSum(xl);
  float rv = waveWmmaSum(xv);
  if (lane == 0) { wq[wave] = rq; wl[wave] = rl; wv[wave] = rv; }
  __syncthreads();

  // Stage B: wave 0 reduces the 8 wave sums (branch is wave-uniform -> EXEC
  // is all ones inside; inactive lanes contribute 0.0f).
  if (wave == 0) {
    const int nw = GNLL_BLOCK / 32;
    float aq = (lane < nw) ? wq[lane] : 0.0f;
    float al = (lane < nw) ? wl[lane] : 0.0f;
    float av = (lane < nw) ? wv[lane] : 0.0f;
    aq = waveWmmaSum(aq);
    al = waveWmmaSum(al);
    av = waveWmmaSum(av);
    if (lane == 0) {
      out[0] = aq;
      out[1] = al;
      out[2] = av;
    }
  }
}

extern "C" void kernel_launch(void* const* d_in, const int* in_sizes, int n_in,
                              void* d_out, int out_size, void* d_ws, size_t ws_size,
                              hipStream_t stream) {
  (void)n_in; (void)out_size;
  const float* prd = (const float*)d_in[0];   // [N,3] f32
  const float* tgt = (const float*)d_in[1];   // [N,3] f32
  const float* fv  = (const float*)d_in[2];   // [N,6] f32
  const int n = in_sizes[0] / 3;

  // Deterministic two-pass reduction through workspace (no atomics).
  int nblocks = GNLL_MAXBLOCKS;
  const size_t need = (size_t)nblocks * 3 * sizeof(double);
  if (ws_size < need) {
    nblocks = (int)(ws_size / (3 * sizeof(double)));
    if (nblocks < 1) nblocks = 1;
  }
  double* partials = (double*)d_ws;

  gnll_partial_kernel<<<nblocks, GNLL_BLOCK, 0, stream>>>(prd, tgt, fv, partials, n);
  gnll_final_kernel<<<1, GNLL_BLOCK, 0, stream>>>(partials, nblocks, (float*)d_out,
                                                  1.0 / (double)n);
}